// GraphSAGE_10282151706738
// MI455X (gfx1250) — compile-verified
//
#include <hip/hip_runtime.h>

typedef __attribute__((ext_vector_type(2))) float v2f;
typedef __attribute__((ext_vector_type(8))) float v8f;

#define N_NODES 100000
#define N_EDGES 1600000
#define D 128
#define COMB_STRIDE 260   // floats per row of interleaved {Wl,Wr} pairs (64 float4 + 1 pad)
#define OUT_STRIDE 132    // floats per row of output staging tile

// ---------------------------------------------------------------------------
// zero fill (float4)
// ---------------------------------------------------------------------------
__global__ void zero_f32(float4* __restrict__ p, long n4) {
    long i = (long)blockIdx.x * blockDim.x + threadIdx.x;
    long stride = (long)gridDim.x * blockDim.x;
    float4 z = {0.f, 0.f, 0.f, 0.f};
    for (; i < n4; i += stride) p[i] = z;
}

// ---------------------------------------------------------------------------
// per-destination degree counts (float, matches reference segment_sum of ones)
// ---------------------------------------------------------------------------
__global__ void edge_count(const int* __restrict__ dst, float* __restrict__ cnt) {
    int e = blockIdx.x * blockDim.x + threadIdx.x;
    if (e < N_EDGES) atomicAdd(&cnt[dst[e]], 1.0f);
}

// ---------------------------------------------------------------------------
// scatter-add of source features into agg[dst]: thread = (edge, 4-feature chunk)
// x rows + agg both L2-resident (51.2 MB each, 192 MB L2) -> atomics at L2.
// ---------------------------------------------------------------------------
__global__ void edge_scatter(const int* __restrict__ src, const int* __restrict__ dst,
                             const float* __restrict__ x, float* __restrict__ agg) {
    long idx = (long)blockIdx.x * blockDim.x + threadIdx.x;
    long e = idx >> 5;                 // 32 chunks of 4 floats per edge
    if (e >= N_EDGES) return;
    int f = (int)(idx & 31) * 4;
    long s = src[e];
    long d = dst[e];
    float4 v = *(const float4*)(x + s * D + f);
    float* ap = agg + d * D + f;
    atomicAdd(ap + 0, v.x);
    atomicAdd(ap + 1, v.y);
    atomicAdd(ap + 2, v.z);
    atomicAdd(ap + 3, v.w);
}

// ---------------------------------------------------------------------------
// Fused: mean = agg / max(cnt,1);  out = mean @ Wl^T + bias + xin @ Wr^T; [relu]
// Block = 256 threads = 8 wave32; block covers 128 node rows; wave w owns a
// 16-row stripe and all 128 output columns (8 x 16x16 f32 C tiles).
// LDS holds Wl/Wr interleaved so ONE ds_load_b128 yields both B fragments.
// After the K loop the weight LDS is reused to transpose-stage the output so
// global stores are coalesced b128 full-line writes.
// ---------------------------------------------------------------------------
__global__ void __launch_bounds__(256)
sage_gemm(const float* __restrict__ agg, const float* __restrict__ cnt,
          const float* __restrict__ xin, const float* __restrict__ Wl,
          const float* __restrict__ bl,  const float* __restrict__ Wr,
          float* __restrict__ out, int relu) {
    extern __shared__ float lds[];   // 128*COMB_STRIDE floats (reused for out-stage)

    // stage interleaved weights: lds[n*COMB_STRIDE + p*4] = {Wl[n][2p],Wl[n][2p+1],
    //                                                        Wr[n][2p],Wr[n][2p+1]}
    for (int i = threadIdx.x; i < D * (D / 2); i += 256) {
        int n = i >> 6;              // D/2 = 64 pairs per row
        int p = i & 63;
        v2f wl = *(const v2f*)(Wl + n * D + 2 * p);
        v2f wr = *(const v2f*)(Wr + n * D + 2 * p);
        float4 c4 = {wl.x, wl.y, wr.x, wr.y};
        *(float4*)(lds + n * COMB_STRIDE + p * 4) = c4;
    }
    __syncthreads();

    const int lane = threadIdx.x & 31;
    const int wave = threadIdx.x >> 5;
    const int half = lane >> 4;      // K-pair select for A/B fragments
    const int l16  = lane & 15;      // M (for A) / N (for B,C) within tile

    // A-fragment row for this lane (clamped for the ragged last block; keeps
    // EXEC all-ones for WMMA, OOB rows handled at the final store)
    long rowA = (long)blockIdx.x * 128 + wave * 16 + l16;
    long rA   = rowA < N_NODES ? rowA : (long)(N_NODES - 1);

    float c   = cnt[rA];
    float inv = 1.0f / (c > 1.0f ? c : 1.0f);

    const float* aggRow = agg + rA * D;
    const float* xRow   = xin + rA * D;

    v8f acc[8] = {};     // 8 column tiles x 8 VGPRs

    for (int k = 0; k < D; k += 4) {
        int ko = k + 2 * half;                       // K offset for this lane half
        int p4 = ((ko >> 1) << 2);                   // float4 index*4 within LDS row
        v2f aM = *(const v2f*)(aggRow + ko);
        aM = aM * inv;                               // on-the-fly mean
        v2f aX = *(const v2f*)(xRow + ko);

        float4 bf[8];                                // prefetch all B fragments
        #pragma unroll
        for (int t = 0; t < 8; ++t) {
            int n = t * 16 + l16;
            bf[t] = *(const float4*)(lds + n * COMB_STRIDE + p4);
        }
        #pragma unroll
        for (int t = 0; t < 8; ++t) {
            v2f bL = {bf[t].x, bf[t].y};             // B = Wl^T fragment
            v2f bR = {bf[t].z, bf[t].w};             // B = Wr^T fragment
            acc[t] = __builtin_amdgcn_wmma_f32_16x16x4_f32(
                false, aM, false, bL, (short)0, acc[t], false, false);
            acc[t] = __builtin_amdgcn_wmma_f32_16x16x4_f32(
                false, aX, false, bR, (short)0, acc[t], false, false);
        }
    }

    // ---- stage results into LDS (bias + relu applied), then coalesced store ----
    __syncthreads();   // weight reads done; safe to reuse LDS

    // C/D layout: VGPR r -> M = r + 8*half ; lane's N = l16 within each tile
    int localBase = wave * 16 + half * 8;
    #pragma unroll
    for (int t = 0; t < 8; ++t) {
        int n = t * 16 + l16;
        float bias = bl[n];
        #pragma unroll
        for (int r = 0; r < 8; ++r) {
            float v = acc[t][r] + bias;
            if (relu) v = v > 0.0f ? v : 0.0f;
            lds[(localBase + r) * OUT_STRIDE + n] = v;
        }
    }
    __syncthreads();

    long blockBase = (long)blockIdx.x * 128;
    for (int i = threadIdx.x; i < 128 * (D / 4); i += 256) {
        int row = i >> 5;            // D/4 = 32 float4 per row
        int c4  = (i & 31) * 4;
        long grow = blockBase + row;
        if (grow < N_NODES) {
            float4 v = *(const float4*)(lds + row * OUT_STRIDE + c4);
            *(float4*)(out + grow * D + c4) = v;
        }
    }
}

// ---------------------------------------------------------------------------
extern "C" void kernel_launch(void* const* d_in, const int* in_sizes, int n_in,
                              void* d_out, int out_size, void* d_ws, size_t ws_size,
                              hipStream_t stream) {
    const float* x   = (const float*)d_in[0];
    const int*   ei  = (const int*)d_in[1];     // [2, E]
    const float* W1l = (const float*)d_in[2];
    const float* b1l = (const float*)d_in[3];
    const float* W1r = (const float*)d_in[4];
    const float* W2l = (const float*)d_in[5];
    const float* b2l = (const float*)d_in[6];
    const float* W2r = (const float*)d_in[7];

    const int* src = ei;
    const int* dst = ei + N_EDGES;

    // workspace layout: agg[N*D] | cnt[N] | h[N*D]   (adjacent -> one zero pass)
    float* agg  = (float*)d_ws;
    float* cntp = agg + (size_t)N_NODES * D;
    float* h    = cntp + N_NODES;
    float* outp = (float*)d_out;

    const long nAgg = (long)N_NODES * D;
    const int  threads = 256;
    const long scatterThreads = (long)N_EDGES * 32;
    const int  scatterBlocks  = (int)((scatterThreads + threads - 1) / threads);
    const int  gemmBlocks     = (N_NODES + 127) / 128;
    const size_t shmem        = (size_t)D * COMB_STRIDE * sizeof(float);  // 133,120 B

    // ---- layer 1 ----
    zero_f32<<<2048, threads, 0, stream>>>((float4*)agg, (nAgg + N_NODES) / 4);  // agg + cnt
    edge_count<<<(N_EDGES + threads - 1) / threads, threads, 0, stream>>>(dst, cntp);
    edge_scatter<<<scatterBlocks, threads, 0, stream>>>(src, dst, x, agg);
    sage_gemm<<<gemmBlocks, threads, shmem, stream>>>(agg, cntp, x, W1l, b1l, W1r, h, 1);

    // ---- layer 2 (counts reused) ----
    zero_f32<<<2048, threads, 0, stream>>>((float4*)agg, nAgg / 4);
    edge_scatter<<<scatterBlocks, threads, 0, stream>>>(src, dst, h, agg);
    sage_gemm<<<gemmBlocks, threads, shmem, stream>>>(agg, cntp, h, W2l, b2l, W2r, outp, 0);
}